// MinibatchDiscrimination_1769526526241
// MI455X (gfx1250) — compile-verified
//
#include <hip/hip_runtime.h>
#include <math.h>

typedef __attribute__((ext_vector_type(2))) float v2f;
typedef __attribute__((ext_vector_type(8))) float v8f;

#define B_DIM 512
#define IN_F  256
#define OUT_F 64
#define KD    16
#define N_DIM (OUT_F * KD)   // 1024
#define OUT_COLS (IN_F + OUT_F) // 320

// ---------------------------------------------------------------------------
// Kernel 1: copy x into out[:, 0:256]
// ---------------------------------------------------------------------------
__global__ void copy_x_kernel(const float* __restrict__ x, float* __restrict__ out) {
    int a = blockIdx.x;          // 0..511
    int i = threadIdx.x;         // 0..255
    out[a * OUT_COLS + i] = x[a * IN_F + i];
}

// ---------------------------------------------------------------------------
// Kernel 2: LDS-tiled transpose T[256,1024] -> Tt[1024,256]
// block (32,8), each block moves a 32x32 tile
// ---------------------------------------------------------------------------
__global__ void transpose_T_kernel(const float* __restrict__ T, float* __restrict__ Tt) {
    __shared__ float tile[32][33];   // +1 pad to kill bank conflicts
    int n0 = blockIdx.x * 32;        // along 1024
    int k0 = blockIdx.y * 32;        // along 256
    #pragma unroll
    for (int i = threadIdx.y; i < 32; i += 8)
        tile[i][threadIdx.x] = T[(k0 + i) * N_DIM + n0 + threadIdx.x];
    __syncthreads();
    #pragma unroll
    for (int i = threadIdx.y; i < 32; i += 8)
        Tt[(n0 + i) * IN_F + k0 + threadIdx.x] = tile[threadIdx.x][i];
}

// ---------------------------------------------------------------------------
// Kernel 3: M = x @ T via V_WMMA_F32_16X16X4_F32 (full f32 matrix pipe).
// One wave per 16x16 output tile; K loop = 64 wmma steps (K=4 each).
// A fragment: lane L -> row (m0 + L%16), K = 4s + 2*(L/16) + {0,1}   (v2f load)
// B fragment: lane L -> col (n0 + L%16) = row of Tt, same K indexing (v2f load)
// C/D: lane L -> col L%16, rows (L/16)*8 + r, r = 0..7
// ---------------------------------------------------------------------------
__global__ void __launch_bounds__(32)
gemm_wmma_f32_kernel(const float* __restrict__ x, const float* __restrict__ Tt,
                     float* __restrict__ M) {
    const int n0   = blockIdx.x * 16;      // 0..1023
    const int m0   = blockIdx.y * 16;      // 0..511
    const int lane = threadIdx.x;          // 0..31
    const int half = lane >> 4;            // 0 or 1
    const int l16  = lane & 15;

    const float* arow = x  + (m0 + l16) * IN_F + 2 * half;
    const float* brow = Tt + (n0 + l16) * IN_F + 2 * half;

    v8f c = {0.f, 0.f, 0.f, 0.f, 0.f, 0.f, 0.f, 0.f};

    #pragma unroll 8
    for (int s = 0; s < IN_F / 4; ++s) {
        v2f a = *(const v2f*)(arow + 4 * s);
        v2f b = *(const v2f*)(brow + 4 * s);
        // 8 args: (neg_a, A, neg_b, B, c_mod, C, reuse_a, reuse_b)
        c = __builtin_amdgcn_wmma_f32_16x16x4_f32(
                false, a, false, b, (short)0, c, false, false);
    }

    float* outp = M + (m0 + half * 8) * N_DIM + n0 + l16;
    #pragma unroll
    for (int r = 0; r < 8; ++r)
        outp[r * N_DIM] = c[r];
}

// ---------------------------------------------------------------------------
// Kernel 4: pairwise L1 -> exp -> masked mean, fused.
// grid (2, 64): blockIdx.y = j (feature), blockIdx.x = a-chunk of 256.
// Each thread owns one 'a' row (16 regs), b-loop tiled 64 rows through LDS.
// ---------------------------------------------------------------------------
__global__ void __launch_bounds__(256)
pairwise_kernel(const float* __restrict__ M, float* __restrict__ out) {
    const int j = blockIdx.y;                        // 0..63
    const int a = blockIdx.x * 256 + threadIdx.x;    // 0..511

    __shared__ float tile[64][KD];                   // 4 KB

    // register-cache this thread's row M[a, j*16 .. j*16+15]
    float ra[KD];
    const float4* rp = (const float4*)(M + a * N_DIM + j * KD);
    #pragma unroll
    for (int q = 0; q < 4; ++q) {
        float4 v = rp[q];
        ra[4 * q + 0] = v.x; ra[4 * q + 1] = v.y;
        ra[4 * q + 2] = v.z; ra[4 * q + 3] = v.w;
    }

    float acc = 0.f;
    for (int bt = 0; bt < B_DIM; bt += 64) {
        // 256 threads load 64x16 floats: one float4 each
        int r  = threadIdx.x >> 2;
        int c4 = (threadIdx.x & 3) * 4;
        *(float4*)&tile[r][c4] =
            *(const float4*)(M + (bt + r) * N_DIM + j * KD + c4);
        __syncthreads();

        #pragma unroll 4
        for (int bb = 0; bb < 64; ++bb) {
            const int b = bt + bb;
            float l1 = 0.f;
            #pragma unroll
            for (int k = 0; k < KD; ++k)
                l1 += fabsf(ra[k] - tile[bb][k]);
            float e = __expf(-l1);
            acc += (b == a) ? 0.f : e;   // zero self-similarity
        }
        __syncthreads();
    }

    // o_b = MINIBATCH_WEIGHT * mean over axis=1 (divide by full B=512)
    out[a * OUT_COLS + IN_F + j] = 0.5f * acc * (1.0f / (float)B_DIM);
}

// ---------------------------------------------------------------------------
// Host launcher
// ---------------------------------------------------------------------------
extern "C" void kernel_launch(void* const* d_in, const int* in_sizes, int n_in,
                              void* d_out, int out_size, void* d_ws, size_t ws_size,
                              hipStream_t stream) {
    const float* x = (const float*)d_in[0];   // [512, 256]
    const float* T = (const float*)d_in[1];   // [256, 64, 16] == [256, 1024]
    float* out = (float*)d_out;               // [512, 320]

    float* Tt = (float*)d_ws;                               // [1024, 256] (1 MB)
    float* M  = (float*)((char*)d_ws + (size_t)N_DIM * IN_F * sizeof(float)); // [512,1024] (2 MB)

    // 1) out[:, :256] = x
    copy_x_kernel<<<dim3(B_DIM), dim3(IN_F), 0, stream>>>(x, out);

    // 2) Tt = T^T  (makes WMMA B-fragment loads contiguous)
    transpose_T_kernel<<<dim3(N_DIM / 32, IN_F / 32), dim3(32, 8), 0, stream>>>(T, Tt);

    // 3) M = x @ T  via f32 WMMA; one wave per 16x16 tile
    gemm_wmma_f32_kernel<<<dim3(N_DIM / 16, B_DIM / 16), dim3(32), 0, stream>>>(x, Tt, M);

    // 4) fused pairwise L1 + exp + masked mean -> out[:, 256:]
    pairwise_kernel<<<dim3(B_DIM / 256, OUT_F), dim3(256), 0, stream>>>(M, out);
}